// GraphAssignAttention_17875653886461
// MI455X (gfx1250) — compile-verified
//
#include <hip/hip_runtime.h>
#include <math.h>

#define B_  4
#define N_  50000
#define C_  128
#define E_  800000
#define S_  64
#define H_  8
#define DH_ 16

typedef __attribute__((ext_vector_type(2))) float v2f;
typedef __attribute__((ext_vector_type(8))) float v8f;

static __device__ __forceinline__ v8f wmma_f32_4(v2f a, v2f b, v8f c) {
  // D = A(16x4) * B(4x16) + C, fp32, wave32
  return __builtin_amdgcn_wmma_f32_16x16x4_f32(false, a, false, b, (short)0, c, false, false);
}

static __device__ __forceinline__ v8f splat8(float v) {
  v8f r = {v, v, v, v, v, v, v, v};
  return r;
}

// ---------------- zero fill ----------------
__global__ void zero_f32(float* __restrict__ p, long n) {
  long i = (long)blockIdx.x * blockDim.x + threadIdx.x;
  long st = (long)gridDim.x * blockDim.x;
  for (; i < n; i += st) p[i] = 0.0f;
}

// ---------------- sparse A @ x scatter (edge parallel) ----------------
__global__ void edge_scatter(const float* __restrict__ x, const int* __restrict__ rows,
                             const int* __restrict__ cols, const float* __restrict__ vals,
                             float* __restrict__ xg) {
  const long total = (long)E_ * 128;  // float4 units per edge payload (B*C/4)
  long t = (long)blockIdx.x * blockDim.x + threadIdx.x;
  long st = (long)gridDim.x * blockDim.x;
  for (; t < total; t += st) {
    int e  = (int)(t >> 7);
    int j  = (int)(t & 127);
    int b  = j >> 5;
    int c4 = j & 31;
    int r = rows[e];
    int c = cols[e];
    float v = vals[e];
    const float4 s = *(const float4*)(x + ((long)b * N_ + c) * C_ + c4 * 4);
    float* d = xg + ((long)b * N_ + r) * C_ + c4 * 4;
    atomicAdd(d + 0, s.x * v);
    atomicAdd(d + 1, s.y * v);
    atomicAdd(d + 2, s.z * v);
    atomicAdd(d + 3, s.w * v);
    long tn = t + st;
    if (tn < total) {  // prefetch next iteration's gather source into cache
      int e2 = (int)(tn >> 7);
      int j2 = (int)(tn & 127);
      __builtin_prefetch(x + ((long)(j2 >> 5) * N_ + cols[e2]) * C_ + (j2 & 31) * 4, 0, 0);
    }
  }
}

// ---------------- fused: H = X Wself^T + XG Wgraph^T + b; LN; GELU; logits; softmax ----------------
__global__ __launch_bounds__(128) void fused_node(
    const float* __restrict__ x, const float* __restrict__ xg,
    const float* __restrict__ w_self, const float* __restrict__ b_self,
    const float* __restrict__ w_graph,
    const float* __restrict__ ln_g, const float* __restrict__ ln_b,
    const float* __restrict__ w_slice, const float* __restrict__ b_slice,
    float* __restrict__ weights) {
  __shared__ float xs[64 * C_];
  __shared__ float xgs[64 * C_];
  const int TPB = (N_ + 63) / 64;
  int b = blockIdx.x / TPB;
  int node0 = (blockIdx.x % TPB) * 64;
  int tid = threadIdx.x;

  // cooperative, coalesced tile load (zeros past N)
  for (int i = tid; i < 64 * C_ / 4; i += 128) {
    int row = i >> 5;
    int c4 = i & 31;
    int node = node0 + row;
    float4 vx = make_float4(0.f, 0.f, 0.f, 0.f);
    float4 vg = vx;
    if (node < N_) {
      long base = ((long)b * N_ + node) * C_ + c4 * 4;
      vx = *(const float4*)(x + base);
      vg = *(const float4*)(xg + base);
    }
    ((float4*)xs)[i]  = vx;
    ((float4*)xgs)[i] = vg;
  }
  __syncthreads();

  int wid = tid >> 5, lane = tid & 31;
  int m = lane & 15, kh = lane >> 4;  // A/B row-col index and K-half per ISA f32 layout
  float* xsl = xs  + wid * 16 * C_;
  float* xgl = xgs + wid * 16 * C_;

  // ---- H (16 x 128) via WMMA f32 16x16x4, accumulators stay in registers ----
  v8f hacc[8];
#pragma unroll
  for (int jt = 0; jt < 8; ++jt) hacc[jt] = splat8(b_self[jt * 16 + m]);
  for (int ks = 0; ks < 32; ++ks) {
    int k0 = ks * 4 + kh * 2;
    v2f a1; a1.x = xsl[m * C_ + k0]; a1.y = xsl[m * C_ + k0 + 1];
    v2f a2; a2.x = xgl[m * C_ + k0]; a2.y = xgl[m * C_ + k0 + 1];
#pragma unroll
    for (int jt = 0; jt < 8; ++jt) {
      const float* wp = w_self + (jt * 16 + m) * C_ + k0;
      v2f b1; b1.x = wp[0]; b1.y = wp[1];
      hacc[jt] = wmma_f32_4(a1, b1, hacc[jt]);
      const float* wg = w_graph + (jt * 16 + m) * C_ + k0;
      v2f b2; b2.x = wg[0]; b2.y = wg[1];
      hacc[jt] = wmma_f32_4(a2, b2, hacc[jt]);
    }
  }
  // write H back into our own xs slice (done reading it)
#pragma unroll
  for (int jt = 0; jt < 8; ++jt)
#pragma unroll
    for (int r = 0; r < 8; ++r)
      xsl[(r + 8 * kh) * C_ + jt * 16 + m] = hacc[jt][r];

  // ---- LayerNorm + exact GELU, one row per lane 0..15 ----
  if (lane < 16) {
    float* hp = xsl + lane * C_;
    float mu = 0.f;
    for (int c = 0; c < C_; ++c) mu += hp[c];
    mu *= (1.0f / C_);
    float var = 0.f;
    for (int c = 0; c < C_; ++c) { float d = hp[c] - mu; var += d * d; }
    var *= (1.0f / C_);
    float inv = rsqrtf(var + 1e-5f);
    for (int c = 0; c < C_; ++c) {
      float hv = (hp[c] - mu) * inv * ln_g[c] + ln_b[c];
      hp[c] = hv * 0.5f * (1.0f + erff(hv * 0.70710678118654752f));
    }
  }

  // ---- logits (16 x 64) = assign @ w_slice^T + b_slice via WMMA ----
  v8f lacc[4];
#pragma unroll
  for (int jt = 0; jt < 4; ++jt) lacc[jt] = splat8(b_slice[jt * 16 + m]);
  for (int ks = 0; ks < 32; ++ks) {
    int k0 = ks * 4 + kh * 2;
    v2f a; a.x = xsl[m * C_ + k0]; a.y = xsl[m * C_ + k0 + 1];
#pragma unroll
    for (int jt = 0; jt < 4; ++jt) {
      const float* wp = w_slice + (jt * 16 + m) * C_ + k0;
      v2f bb; bb.x = wp[0]; bb.y = wp[1];
      lacc[jt] = wmma_f32_4(a, bb, lacc[jt]);
    }
  }
  float* lg = xgl;  // reuse our xg slice (16*128 >= 16*64)
#pragma unroll
  for (int jt = 0; jt < 4; ++jt)
#pragma unroll
    for (int r = 0; r < 8; ++r)
      lg[(r + 8 * kh) * S_ + jt * 16 + m] = lacc[jt][r];

  // ---- softmax per row over S=64 ----
  if (lane < 16) {
    float* lp = lg + lane * S_;
    float mx = lp[0];
    for (int s = 1; s < S_; ++s) mx = fmaxf(mx, lp[s]);
    float sum = 0.f;
    for (int s = 0; s < S_; ++s) { float e = __expf(lp[s] - mx); lp[s] = e; sum += e; }
    float inv = 1.0f / sum;
    for (int s = 0; s < S_; ++s) lp[s] *= inv;
  }

  // ---- coalesced store of weights (B,N,S) ----
  int wnode0 = node0 + wid * 16;
  for (int i = lane; i < 16 * S_; i += 32) {
    int r = i >> 6, s = i & 63;
    int node = wnode0 + r;
    if (node < N_) weights[((long)b * N_ + node) * S_ + s] = lg[i];
  }
}

// ---------------- wsum[b,s] = max(sum_n weights, 1e-8) ----------------
__global__ __launch_bounds__(256) void wsum_kernel(const float* __restrict__ weights,
                                                   float* __restrict__ wsum) {
  __shared__ float red[256];
  int b = blockIdx.x / S_, s = blockIdx.x % S_;
  float p = 0.f;
  for (int n = threadIdx.x; n < N_; n += 256)
    p += weights[((long)b * N_ + n) * S_ + s];
  red[threadIdx.x] = p;
  __syncthreads();
  for (int o = 128; o > 0; o >>= 1) {
    if (threadIdx.x < o) red[threadIdx.x] += red[threadIdx.x + o];
    __syncthreads();
  }
  if (threadIdx.x == 0) wsum[blockIdx.x] = fmaxf(red[0], 1e-8f);
}

// ---------------- slices[b,s,c] += sum_n W[b,n,s] x[b,n,c] (WMMA split-K + atomics) ----------------
__global__ __launch_bounds__(32) void slices_partial(const float* __restrict__ weights,
                                                     const float* __restrict__ x,
                                                     float* __restrict__ slices) {
  const int NCH = (N_ + 511) / 512;
  int bidx = blockIdx.x;
  int b = bidx / (4 * NCH);
  int rem = bidx % (4 * NCH);
  int st = rem / NCH;
  int ch = rem % NCH;
  int k_beg = ch * 512;
  int k_end = k_beg + 512; if (k_end > N_) k_end = N_;
  int lane = threadIdx.x, m = lane & 15, kh = lane >> 4;
  v8f acc[8] = {};
  for (int k = k_beg; k < k_end; k += 4) {
    int k0 = k + kh * 2;
    v2f a;
    a.x = weights[((long)b * N_ + k0) * S_ + st * 16 + m];
    a.y = weights[((long)b * N_ + k0 + 1) * S_ + st * 16 + m];
    const float* xr0 = x + ((long)b * N_ + k0) * C_;
    const float* xr1 = xr0 + C_;
#pragma unroll
    for (int ct = 0; ct < 8; ++ct) {
      v2f bb; bb.x = xr0[ct * 16 + m]; bb.y = xr1[ct * 16 + m];
      acc[ct] = wmma_f32_4(a, bb, acc[ct]);
    }
  }
#pragma unroll
  for (int ct = 0; ct < 8; ++ct)
#pragma unroll
    for (int r = 0; r < 8; ++r)
      atomicAdd(&slices[((long)b * S_ + st * 16 + r + 8 * kh) * C_ + ct * 16 + m], acc[ct][r]);
}

// ---------------- tiny kernels: normalize, qkv, attention, out-proj ----------------
__global__ void slices_norm_k(const float* __restrict__ slices, const float* __restrict__ wsum,
                              float* __restrict__ sn) {
  int i = blockIdx.x * blockDim.x + threadIdx.x;
  if (i < B_ * S_ * C_) sn[i] = slices[i] / wsum[i / C_];
}

__global__ void qkv_k(const float* __restrict__ sn, const float* __restrict__ w,
                      const float* __restrict__ bias, float* __restrict__ qkv) {
  int i = blockIdx.x * blockDim.x + threadIdx.x;
  if (i >= B_ * S_ * 3 * C_) return;
  int j = i % (3 * C_);
  int bs = i / (3 * C_);
  float acc = bias[j];
  const float* xr = sn + (long)bs * C_;
  const float* wr = w + (long)j * C_;
  for (int c = 0; c < C_; ++c) acc += xr[c] * wr[c];
  qkv[i] = acc;
}

__global__ __launch_bounds__(64) void attn_k(const float* __restrict__ qkv, float* __restrict__ o) {
  int b = blockIdx.x / H_, h = blockIdx.x % H_;
  int q = threadIdx.x;
  const float* Q = qkv + (long)(b * S_ + q) * (3 * C_) + h * DH_;
  float sc[S_];
  float mx = -1e30f;
  for (int k = 0; k < S_; ++k) {
    const float* K = qkv + (long)(b * S_ + k) * (3 * C_) + C_ + h * DH_;
    float d = 0.f;
    for (int t = 0; t < DH_; ++t) d += Q[t] * K[t];
    d *= 0.25f;  // 1/sqrt(16)
    sc[k] = d;
    mx = fmaxf(mx, d);
  }
  float sum = 0.f;
  for (int k = 0; k < S_; ++k) { float e = __expf(sc[k] - mx); sc[k] = e; sum += e; }
  float inv = 1.0f / sum;
  float ov[DH_];
  for (int t = 0; t < DH_; ++t) ov[t] = 0.f;
  for (int k = 0; k < S_; ++k) {
    const float* V = qkv + (long)(b * S_ + k) * (3 * C_) + 2 * C_ + h * DH_;
    float p = sc[k] * inv;
    for (int t = 0; t < DH_; ++t) ov[t] += p * V[t];
  }
  for (int t = 0; t < DH_; ++t) o[(long)(b * S_ + q) * C_ + h * DH_ + t] = ov[t];
}

__global__ void outproj_k(const float* __restrict__ o, const float* __restrict__ w,
                          const float* __restrict__ bias, float* __restrict__ so) {
  int i = blockIdx.x * blockDim.x + threadIdx.x;
  if (i >= B_ * S_ * C_) return;
  int c = i % C_;
  int bs = i / C_;
  float acc = bias[c];
  const float* orow = o + (long)bs * C_;
  const float* wrow = w + (long)c * C_;
  for (int k = 0; k < C_; ++k) acc += orow[k] * wrow[k];
  so[i] = acc;
}

// ---------------- out[b,n,c] = sum_s W[b,n,s] * slices_out[b,s,c] (WMMA, K=64) ----------------
__global__ __launch_bounds__(32) void expand_k(const float* __restrict__ weights,
                                               const float* __restrict__ so,
                                               float* __restrict__ out) {
  __shared__ float sh[16 * C_];
  const int NT = N_ / 16;  // N divisible by 16
  int b = blockIdx.x / NT;
  int node0 = (blockIdx.x % NT) * 16;
  int lane = threadIdx.x, m = lane & 15, kh = lane >> 4;
  v8f acc[8] = {};
  for (int ks = 0; ks < 16; ++ks) {
    int k0 = ks * 4 + kh * 2;
    const float* wr = weights + ((long)b * N_ + node0 + m) * S_ + k0;
    v2f a; a.x = wr[0]; a.y = wr[1];
    const float* s0 = so + ((long)b * S_ + k0) * C_;
    const float* s1 = s0 + C_;
#pragma unroll
    for (int ct = 0; ct < 8; ++ct) {
      v2f bb; bb.x = s0[ct * 16 + m]; bb.y = s1[ct * 16 + m];
      acc[ct] = wmma_f32_4(a, bb, acc[ct]);
    }
  }
#pragma unroll
  for (int ct = 0; ct < 8; ++ct)
#pragma unroll
    for (int r = 0; r < 8; ++r)
      sh[(r + 8 * kh) * C_ + ct * 16 + m] = acc[ct][r];
  // single wave: DS ops in-order; coalesced 128B stores
  for (int i = lane; i < 16 * C_ / 4; i += 32) {
    int row = i >> 5, c4 = i & 31;
    ((float4*)(out + ((long)b * N_ + node0 + row) * C_))[c4] = ((float4*)sh)[i];
  }
}

extern "C" void kernel_launch(void* const* d_in, const int* in_sizes, int n_in,
                              void* d_out, int out_size, void* d_ws, size_t ws_size,
                              hipStream_t stream) {
  (void)in_sizes; (void)n_in; (void)out_size; (void)ws_size;
  const float* x       = (const float*)d_in[0];
  const int*   adj     = (const int*)d_in[1];
  const float* vals    = (const float*)d_in[2];
  const float* w_self  = (const float*)d_in[3];
  const float* b_self  = (const float*)d_in[4];
  const float* w_graph = (const float*)d_in[5];
  const float* ln_g    = (const float*)d_in[6];
  const float* ln_b    = (const float*)d_in[7];
  const float* w_slice = (const float*)d_in[8];
  const float* b_slice = (const float*)d_in[9];
  const float* ipw     = (const float*)d_in[10];
  const float* ipb     = (const float*)d_in[11];
  const float* opw     = (const float*)d_in[12];
  const float* opb     = (const float*)d_in[13];
  float* out = (float*)d_out;

  float* ws      = (float*)d_ws;
  float* xg      = ws;                               // B*N*C
  float* weights = xg + (long)B_ * N_ * C_;          // B*N*S
  float* slices  = weights + (long)B_ * N_ * S_;     // B*S*C
  float* wsum    = slices + B_ * S_ * C_;            // B*S
  float* sn      = wsum + B_ * S_;                   // B*S*C
  float* qkv     = sn + B_ * S_ * C_;                // B*S*3C
  float* ao      = qkv + B_ * S_ * 3 * C_;           // B*S*C
  float* so      = ao + B_ * S_ * C_;                // B*S*C

  const int* rows = adj;
  const int* cols = adj + E_;

  zero_f32<<<4096, 256, 0, stream>>>(xg, (long)B_ * N_ * C_);
  zero_f32<<<64, 256, 0, stream>>>(slices, (long)B_ * S_ * C_);
  edge_scatter<<<16384, 256, 0, stream>>>(x, rows, cols, vals, xg);

  const int TPB = (N_ + 63) / 64;
  fused_node<<<B_ * TPB, 128, 0, stream>>>(x, xg, w_self, b_self, w_graph,
                                           ln_g, ln_b, w_slice, b_slice, weights);

  wsum_kernel<<<B_ * S_, 256, 0, stream>>>(weights, wsum);

  const int NCH = (N_ + 511) / 512;
  slices_partial<<<B_ * 4 * NCH, 32, 0, stream>>>(weights, x, slices);

  slices_norm_k<<<(B_ * S_ * C_ + 255) / 256, 256, 0, stream>>>(slices, wsum, sn);
  qkv_k<<<(B_ * S_ * 3 * C_ + 255) / 256, 256, 0, stream>>>(sn, ipw, ipb, qkv);
  attn_k<<<B_ * H_, 64, 0, stream>>>(qkv, ao);
  outproj_k<<<(B_ * S_ * C_ + 255) / 256, 256, 0, stream>>>(ao, opw, opb, so);

  expand_k<<<B_ * (N_ / 16), 32, 0, stream>>>(weights, so, out);
}